// ScaledDotProductAttention_8478265442343
// MI455X (gfx1250) — compile-verified
//
#include <hip/hip_runtime.h>
#include <stdint.h>

typedef __attribute__((ext_vector_type(2))) float v2f;
typedef __attribute__((ext_vector_type(4))) float v4f;
typedef __attribute__((ext_vector_type(8))) float v8f;

#define BH_   64
#define TQ_   1024
#define TK_   1024
#define DK_   64
#define WAVES 4
#define NEG_MAX (-3.402823466e38f)

// Fused masked-softmax attention, one 16-row Q tile per workgroup.
// Phase 1: S = Q K^T / sqrt(dk) via v_wmma_f32_16x16x4_f32 -> LDS strip [16 x TK]
// Phase 2: masked softmax (+ fully-masked zeroing + head ablation) in LDS,
//          vectorized b128 path, weights streamed to HBM non-temporally.
// Phase 3: O = W V via v_wmma_f32_16x16x4_f32, W read from LDS (no HBM re-read).
__global__ __launch_bounds__(128) void sdpa_wmma_f32_kernel(
    const float* __restrict__ Q, const float* __restrict__ K,
    const float* __restrict__ V, const uint8_t* __restrict__ mask,
    const int* __restrict__ p_head_ablation, const int* __restrict__ p_num_heads,
    float* __restrict__ out,      // [BH, TQ, DK]
    float* __restrict__ weights)  // [BH, TQ, TK]
{
    __shared__ float sW[16 * TK_];   // 64 KB: scores, then weights, for this q-tile

    const int bh   = blockIdx.y;
    const int q0   = blockIdx.x * 16;
    const int lane = threadIdx.x & 31;
    const int wave = threadIdx.x >> 5;
    const int hsel = lane >> 4;      // 0: K={0,1}, 1: K={2,3} half of the wave
    const int l15  = lane & 15;

    const float* Qb = Q + ((size_t)bh * TQ_ + q0) * DK_;
    const float* Kb = K + (size_t)bh * TK_ * DK_;
    const float* Vb = V + (size_t)bh * TK_ * DK_;

    // ---------------- Phase 1: scores into LDS ----------------
    // A layout (16x4 f32): lane<16 -> (M=lane, k=d+{0,1}); lane>=16 -> (M=lane-16, k=d+{2,3}).
    v2f qa[DK_ / 4];
#pragma unroll
    for (int d4 = 0; d4 < DK_ / 4; ++d4) {
        const float* p = Qb + (size_t)l15 * DK_ + d4 * 4 + hsel * 2;
        v2f a; a.x = p[0]; a.y = p[1];
        qa[d4] = a;
    }

    for (int kt = wave; kt < TK_ / 16; kt += WAVES) {
        const int k0 = kt * 16;
        v8f c = {};
#pragma unroll
        for (int d4 = 0; d4 < DK_ / 4; ++d4) {
            // B[r][n] = K[k0+n][4*d4+r]; same lane mapping as A.
            const float* p = Kb + (size_t)(k0 + l15) * DK_ + d4 * 4 + hsel * 2;
            v2f b; b.x = p[0]; b.y = p[1];
            c = __builtin_amdgcn_wmma_f32_16x16x4_f32(
                    false, qa[d4], false, b, (short)0, c, false, false);
        }
        // C/D layout: vgpr i -> M = i + 8*hsel, N = l15. Scale by 1/sqrt(64).
#pragma unroll
        for (int i = 0; i < 8; ++i) {
            const int row = i + 8 * hsel;
            sW[row * TK_ + k0 + l15] = c[i] * 0.125f;
        }
    }
    __syncthreads();

    // ---------------- Phase 2: masked softmax + weights writeback ----------------
    const int  nh     = p_num_heads[0];
    const int  ha     = p_head_ablation[0];
    const bool ablate = ((bh % nh) == ha);
    const uint8_t* mrow = mask + (size_t)bh * TK_;

    // Mask is row-invariant: load once per wave (4 bytes/lane/pass) and reduce
    // the fully-masked predicate once.
    uint32_t mw[TK_ / 128];
    int lvalid = 0;
#pragma unroll
    for (int j = 0; j < TK_ / 128; ++j) {
        const uint32_t m4 = *(const uint32_t*)(mrow + j * 128 + lane * 4);
        mw[j] = m4;
        lvalid += ((m4 & 0x000000FFu) ? 1 : 0) + ((m4 & 0x0000FF00u) ? 1 : 0)
                + ((m4 & 0x00FF0000u) ? 1 : 0) + ((m4 & 0xFF000000u) ? 1 : 0);
    }
#pragma unroll
    for (int off = 16; off > 0; off >>= 1)
        lvalid += __shfl_xor(lvalid, off, 32);
    const bool zero_row = (lvalid == 0) || ablate;

    for (int r = wave; r < 16; r += WAVES) {
        float* srow = sW + r * TK_;
        v4f s[TK_ / 128];
        float lmax = NEG_MAX;
#pragma unroll
        for (int j = 0; j < TK_ / 128; ++j) {
            v4f sv = *(const v4f*)(srow + j * 128 + lane * 4);
            const uint32_t m4 = mw[j];
            sv.x = (m4 & 0x000000FFu) ? sv.x : NEG_MAX;
            sv.y = (m4 & 0x0000FF00u) ? sv.y : NEG_MAX;
            sv.z = (m4 & 0x00FF0000u) ? sv.z : NEG_MAX;
            sv.w = (m4 & 0xFF000000u) ? sv.w : NEG_MAX;
            s[j] = sv;
            lmax = fmaxf(lmax, fmaxf(fmaxf(sv.x, sv.y), fmaxf(sv.z, sv.w)));
        }
#pragma unroll
        for (int off = 16; off > 0; off >>= 1)
            lmax = fmaxf(lmax, __shfl_xor(lmax, off, 32));

        float lsum = 0.f;
#pragma unroll
        for (int j = 0; j < TK_ / 128; ++j) {
            v4f ev;
            ev.x = __expf(s[j].x - lmax);
            ev.y = __expf(s[j].y - lmax);
            ev.z = __expf(s[j].z - lmax);
            ev.w = __expf(s[j].w - lmax);
            s[j] = ev;
            lsum += (ev.x + ev.y) + (ev.z + ev.w);
        }
#pragma unroll
        for (int off = 16; off > 0; off >>= 1)
            lsum += __shfl_xor(lsum, off, 32);

        const float scale = zero_row ? 0.f : (1.f / lsum);
        float* wrow = weights + ((size_t)bh * TQ_ + q0 + r) * TK_;
#pragma unroll
        for (int j = 0; j < TK_ / 128; ++j) {
            v4f wv = s[j] * scale;
            *(v4f*)(srow + j * 128 + lane * 4) = wv;           // phase-3 operand
            // 268 MB streamed once, never re-read on-chip -> non-temporal store
            __builtin_nontemporal_store(wv, (v4f*)(wrow + j * 128 + lane * 4));
        }
    }
    __syncthreads();

    // ---------------- Phase 3: O = W V, one 16x16 N-tile per wave ----------------
    const int n0 = wave * 16;     // DK = 64 -> 4 tiles, 4 waves
    v8f o = {};
    for (int k4 = 0; k4 < TK_ / 4; ++k4) {
        const int kb = k4 * 4;
        // A[m][r] = W[q0+m][kb+r] from LDS
        const float* ap = sW + (size_t)l15 * TK_ + kb + hsel * 2;
        v2f a; a.x = ap[0]; a.y = ap[1];
        // B[r][n] = V[kb+r][n0+n]
        const float* bp = Vb + (size_t)(kb + hsel * 2) * DK_ + n0 + l15;
        v2f b; b.x = bp[0]; b.y = bp[DK_];
        o = __builtin_amdgcn_wmma_f32_16x16x4_f32(
                false, a, false, b, (short)0, o, false, false);
    }
    float* obase = out + ((size_t)bh * TQ_ + q0) * DK_;
#pragma unroll
    for (int i = 0; i < 8; ++i) {
        const int row = i + 8 * hsel;
        obase[row * DK_ + n0 + l15] = o[i];
    }
}

extern "C" void kernel_launch(void* const* d_in, const int* in_sizes, int n_in,
                              void* d_out, int out_size, void* d_ws, size_t ws_size,
                              hipStream_t stream) {
    (void)in_sizes; (void)n_in; (void)out_size; (void)d_ws; (void)ws_size;
    const float*   Q    = (const float*)d_in[0];
    const float*   K    = (const float*)d_in[1];
    const float*   V    = (const float*)d_in[2];
    const uint8_t* mask = (const uint8_t*)d_in[3];   // jnp.bool_ -> 1 byte/elem
    const int*     p_ha = (const int*)d_in[4];
    const int*     p_nh = (const int*)d_in[5];

    float* out     = (float*)d_out;                       // [BH,TQ,DK] first
    float* weights = out + (size_t)BH_ * TQ_ * DK_;       // then [BH,TQ,TK]

    dim3 grid(TQ_ / 16, BH_);
    dim3 block(128);
    sdpa_wmma_f32_kernel<<<grid, block, 0, stream>>>(
        Q, K, V, mask, p_ha, p_nh, out, weights);
}